// LRUFullParallel_80161269613383
// MI455X (gfx1250) — compile-verified
//
#include <hip/hip_runtime.h>
#include <hip/hip_bf16.h>
#include <stdint.h>

// ---------------- problem constants ----------------
#define BATCH 8
#define T 4096
#define IN 128
#define OUT 128
#define HID 256
#define H2 (2 * HID)        // 512: re||im packed hidden width
#define L 64                // timestep chunk per workgroup
#define NC (T / L)          // 64 chunks

// LDS tile strides (padded for bank-conflict-free fragment loads)
#define XS_STRIDE 130       // X tile [L][XS_STRIDE]
#define BXS_STRIDE 516      // BX/hidden tile [L][BXS_STRIDE]

// ---------------- workspace layout (floats) ----------------
// W    [H2][IN]   gamma-scaled B (re rows 0..255, im rows 256..511)
// CC   [OUT][H2]  [C_re | -C_im]
// lam  [HID][2]
// lamL [HID][2]   lam^L
// S    [BATCH][NC][HID][2]  per-chunk local final state
// CAR  [BATCH][NC][HID][2]  carry-in state per chunk
#define W_OFF    0
#define CC_OFF   (W_OFF + H2 * IN)          // 65536
#define LAM_OFF  (CC_OFF + OUT * H2)        // 131072
#define LAML_OFF (LAM_OFF + HID * 2)        // 131584
#define S_OFF    (LAML_OFF + HID * 2)       // 132096
#define CAR_OFF  (S_OFF + BATCH * NC * HID * 2)  // 394240
// total = 656384 floats ~= 2.6 MB of d_ws

typedef __attribute__((ext_vector_type(2))) float v2f;
typedef __attribute__((ext_vector_type(8))) float v8f;

static __device__ __forceinline__ v8f wmma_f32(v2f a, v2f b, v8f c) {
  // V_WMMA_F32_16X16X4_F32: D = A(16x4) * B(4x16) + C(16x16), fp32 throughout
  return __builtin_amdgcn_wmma_f32_16x16x4_f32(
      /*neg_a=*/false, a, /*neg_b=*/false, b,
      /*c_mod=*/(short)0, c, /*reuse_a=*/false, /*reuse_b=*/false);
}

static __device__ __forceinline__ v8f zero8() {
  v8f z = {0.f, 0.f, 0.f, 0.f, 0.f, 0.f, 0.f, 0.f};
  return z;
}

// ---------------- kernel 0: parameter prep ----------------
__global__ __launch_bounds__(256) void lru_prep_kernel(
    const float* __restrict__ nu_log, const float* __restrict__ theta_log,
    const float* __restrict__ B_re, const float* __restrict__ B_im,
    const float* __restrict__ C_re, const float* __restrict__ C_im,
    float* __restrict__ ws) {
  __shared__ float gamma_s[HID];
  const int tid = threadIdx.x;  // 0..255, one hidden channel each

  const float nu  = expf(nu_log[tid]);
  const float mag = expf(-nu);
  const float th  = expf(theta_log[tid]);
  const float lre = mag * cosf(th);
  const float lim = mag * sinf(th);
  gamma_s[tid] = sqrtf(fmaxf(0.f, 1.f - mag * mag));

  if (blockIdx.x == 0) {
    ws[LAM_OFF + 2 * tid]     = lre;
    ws[LAM_OFF + 2 * tid + 1] = lim;
    float ar = lre, ai = lim;
#pragma unroll
    for (int i = 0; i < 6; ++i) {  // lam^64 via 6 complex squarings
      const float nr = ar * ar - ai * ai;
      const float ni = 2.f * ar * ai;
      ar = nr; ai = ni;
    }
    ws[LAML_OFF + 2 * tid]     = ar;
    ws[LAML_OFF + 2 * tid + 1] = ai;
  }
  __syncthreads();

  // W[r][i] = gamma[h] * (r<HID ? B_re[h][i] : B_im[h][i]),  h = r % HID
  const int stride = gridDim.x * 256;
  for (int idx = blockIdx.x * 256 + tid; idx < H2 * IN; idx += stride) {
    const int r = idx / IN, i = idx % IN;
    const int h = (r < HID) ? r : (r - HID);
    const float src = (r < HID) ? B_re[h * IN + i] : B_im[h * IN + i];
    ws[W_OFF + idx] = gamma_s[h] * src;
  }
  // CC[o][j] = j<HID ? C_re[o][j] : -C_im[o][j-HID]
  for (int idx = blockIdx.x * 256 + tid; idx < OUT * H2; idx += stride) {
    const int o = idx / H2, j = idx % H2;
    ws[CC_OFF + idx] = (j < HID) ? C_re[o * HID + j] : -C_im[o * HID + (j - HID)];
  }
}

// ---------------- shared device helpers ----------------
// Async global->LDS copy of the X tile (GLOBAL_LOAD_ASYNC_TO_LDS_B128,
// tracked by ASYNCcnt): 32 KB moved without a VGPR round trip.
static __device__ __forceinline__ void load_x_tile_async(
    const float* __restrict__ Xt, float* __restrict__ xs, int tid) {
  // addrspace(3) offset == low 32 bits of the flat LDS pointer (aperture
  // truncation, ISA 10.2); hardware adds LDS_BASE.
  const uint32_t xs_base = (uint32_t)(uintptr_t)xs;
#pragma unroll
  for (int q = tid; q < (L * IN) / 4; q += 256) {  // 8 x b128 per thread
    const int e = q * 4;
    const int row = e / IN, col = e % IN;           // 16B chunks never span rows
    const uint32_t lds_addr = xs_base + (uint32_t)(row * XS_STRIDE + col) * 4u;
    const uint64_t gaddr = (uint64_t)(uintptr_t)(Xt + e);
    asm volatile("global_load_async_to_lds_b128 %0, %1, off"
                 :: "v"(lds_addr), "v"(gaddr)
                 : "memory");
  }
  asm volatile("s_wait_asynccnt 0" ::: "memory");
}

// BX[64][512] = Xtile[64][128] @ W^T[128][512]; W row-major [H2][IN] in global.
static __device__ __forceinline__ void gemm_bx(
    const float* __restrict__ xs, float* __restrict__ bxs,
    const float* __restrict__ Wp, int wave, int lo, int hi) {
  for (int nt = wave * 4; nt < wave * 4 + 4; ++nt) {  // 32 n-tiles / 8 waves
    const int n0 = nt * 16;
    v8f acc[4];
#pragma unroll
    for (int mi = 0; mi < 4; ++mi) acc[mi] = zero8();
    const float* wrow = Wp + (n0 + lo) * IN + 2 * hi;
#pragma unroll 4
    for (int k = 0; k < IN / 4; ++k) {
      const int koff = k * 4;
      const v2f bfrag = *(const v2f*)(wrow + koff);
#pragma unroll
      for (int mi = 0; mi < 4; ++mi) {
        const v2f afrag = *(const v2f*)(xs + (mi * 16 + lo) * XS_STRIDE + koff + 2 * hi);
        acc[mi] = wmma_f32(afrag, bfrag, acc[mi]);
      }
    }
#pragma unroll
    for (int mi = 0; mi < 4; ++mi)
#pragma unroll
      for (int r = 0; r < 8; ++r)
        bxs[(mi * 16 + r + 8 * hi) * BXS_STRIDE + n0 + lo] = acc[mi][r];
  }
}

// ---------------- kernel A: per-chunk local scan final state ----------------
__global__ __launch_bounds__(256) void lru_chunk_state_kernel(
    const float* __restrict__ X, float* __restrict__ ws) {
  extern __shared__ float smem[];
  float* xs  = smem;                  // [L][XS_STRIDE]
  float* bxs = smem + L * XS_STRIDE;  // [L][BXS_STRIDE]

  const int c = blockIdx.x, b = blockIdx.y;
  const int tid = threadIdx.x;
  const int lane = tid & 31, wave = tid >> 5;
  const int lo = lane & 15, hi = lane >> 4;

  load_x_tile_async(X + ((size_t)b * T + (size_t)c * L) * IN, xs, tid);
  __syncthreads();
  gemm_bx(xs, bxs, ws + W_OFF, wave, lo, hi);
  __syncthreads();

  // local diagonal complex scan from zero state; thread = hidden channel
  const int h = tid;
  const float lre = ws[LAM_OFF + 2 * h];
  const float lim = ws[LAM_OFF + 2 * h + 1];
  float sre = 0.f, sim = 0.f;
#pragma unroll 4
  for (int t = 0; t < L; ++t) {
    const float xre = bxs[t * BXS_STRIDE + h];
    const float xim = bxs[t * BXS_STRIDE + HID + h];
    const float nre = fmaf(lre, sre, fmaf(-lim, sim, xre));
    const float nim = fmaf(lre, sim, fmaf(lim, sre, xim));
    sre = nre; sim = nim;
  }
  float* Sp = ws + S_OFF + (((size_t)b * NC + c) * HID + h) * 2;
  Sp[0] = sre; Sp[1] = sim;
}

// ---------------- kernel B: prefix over chunk states ----------------
__global__ __launch_bounds__(256) void lru_carry_kernel(float* __restrict__ ws) {
  const int b = blockIdx.x;       // 0..7
  const int h = threadIdx.x;      // 0..255
  const float Lre = ws[LAML_OFF + 2 * h];
  const float Lim = ws[LAML_OFF + 2 * h + 1];
  const float* S = ws + S_OFF + (size_t)b * NC * HID * 2;
  float* C = ws + CAR_OFF + (size_t)b * NC * HID * 2;
  float sre = 0.f, sim = 0.f;
  C[h * 2] = 0.f; C[h * 2 + 1] = 0.f;
  for (int c = 1; c < NC; ++c) {
    const float pr = S[((c - 1) * HID + h) * 2];
    const float pi = S[((c - 1) * HID + h) * 2 + 1];
    const float nr = fmaf(Lre, sre, fmaf(-Lim, sim, pr));
    const float ni = fmaf(Lre, sim, fmaf(Lim, sre, pi));
    sre = nr; sim = ni;
    C[(c * HID + h) * 2]     = sre;
    C[(c * HID + h) * 2 + 1] = sim;
  }
}

// ---------------- kernel C: exact scan + fused output GEMM ----------------
__global__ __launch_bounds__(256) void lru_final_kernel(
    const float* __restrict__ X, const float* __restrict__ Dw,
    const float* __restrict__ ws, float* __restrict__ Y) {
  extern __shared__ float smem[];
  float* xs  = smem;
  float* bxs = smem + L * XS_STRIDE;

  const int c = blockIdx.x, b = blockIdx.y;
  const int tid = threadIdx.x;
  const int lane = tid & 31, wave = tid >> 5;
  const int lo = lane & 15, hi = lane >> 4;

  load_x_tile_async(X + ((size_t)b * T + (size_t)c * L) * IN, xs, tid);
  __syncthreads();
  gemm_bx(xs, bxs, ws + W_OFF, wave, lo, hi);
  __syncthreads();

  // exact scan seeded with carry-in; rewrite hidden states into bxs in place
  {
    const int h = tid;
    const float lre = ws[LAM_OFF + 2 * h];
    const float lim = ws[LAM_OFF + 2 * h + 1];
    const float* Cp = ws + CAR_OFF + (((size_t)b * NC + c) * HID + h) * 2;
    float sre = Cp[0], sim = Cp[1];
#pragma unroll 4
    for (int t = 0; t < L; ++t) {
      const float xre = bxs[t * BXS_STRIDE + h];
      const float xim = bxs[t * BXS_STRIDE + HID + h];
      const float nre = fmaf(lre, sre, fmaf(-lim, sim, xre));
      const float nim = fmaf(lre, sim, fmaf(lim, sre, xim));
      sre = nre; sim = nim;
      bxs[t * BXS_STRIDE + h]       = sre;
      bxs[t * BXS_STRIDE + HID + h] = sim;
    }
  }
  __syncthreads();

  // Y tile [64][128] = H[64][512] @ CC^T + Xtile[64][128] @ D^T
  const float* CCp = ws + CC_OFF;
  const int n0 = wave * 16;  // 8 n-tiles, one per wave
  v8f acc[4];
#pragma unroll
  for (int mi = 0; mi < 4; ++mi) acc[mi] = zero8();

  const float* ccrow = CCp + (n0 + lo) * H2 + 2 * hi;
#pragma unroll 4
  for (int k = 0; k < H2 / 4; ++k) {  // hidden contraction, K=512
    const int koff = k * 4;
    const v2f bfrag = *(const v2f*)(ccrow + koff);
#pragma unroll
    for (int mi = 0; mi < 4; ++mi) {
      const v2f afrag = *(const v2f*)(bxs + (mi * 16 + lo) * BXS_STRIDE + koff + 2 * hi);
      acc[mi] = wmma_f32(afrag, bfrag, acc[mi]);
    }
  }
  const float* drow = Dw + (n0 + lo) * IN + 2 * hi;
#pragma unroll 4
  for (int k = 0; k < IN / 4; ++k) {  // X @ D^T, K=128
    const int koff = k * 4;
    const v2f bfrag = *(const v2f*)(drow + koff);
#pragma unroll
    for (int mi = 0; mi < 4; ++mi) {
      const v2f afrag = *(const v2f*)(xs + (mi * 16 + lo) * XS_STRIDE + koff + 2 * hi);
      acc[mi] = wmma_f32(afrag, bfrag, acc[mi]);
    }
  }

  float* Yt = Y + ((size_t)b * T + (size_t)c * L) * OUT;
#pragma unroll
  for (int mi = 0; mi < 4; ++mi)
#pragma unroll
    for (int r = 0; r < 8; ++r)
      Yt[(size_t)(mi * 16 + r + 8 * hi) * OUT + n0 + lo] = acc[mi][r];
}

// ---------------- host launcher ----------------
extern "C" void kernel_launch(void* const* d_in, const int* in_sizes, int n_in,
                              void* d_out, int out_size, void* d_ws, size_t ws_size,
                              hipStream_t stream) {
  const float* X         = (const float*)d_in[0];
  const float* nu_log    = (const float*)d_in[1];
  const float* theta_log = (const float*)d_in[2];
  const float* B_re      = (const float*)d_in[3];
  const float* B_im      = (const float*)d_in[4];
  const float* C_re      = (const float*)d_in[5];
  const float* C_im      = (const float*)d_in[6];
  const float* Dw        = (const float*)d_in[7];
  float* Y  = (float*)d_out;
  float* ws = (float*)d_ws;   // needs ~2.6 MB

  const size_t lds_bytes = (size_t)(L * XS_STRIDE + L * BXS_STRIDE) * sizeof(float);

  lru_prep_kernel<<<64, 256, 0, stream>>>(nu_log, theta_log, B_re, B_im, C_re, C_im, ws);
  lru_chunk_state_kernel<<<dim3(NC, BATCH), 256, lds_bytes, stream>>>(X, ws);
  lru_carry_kernel<<<BATCH, 256, 0, stream>>>(ws);
  lru_final_kernel<<<dim3(NC, BATCH), 256, lds_bytes, stream>>>(X, Dw, ws, Y);
}